// GCN_9242769622550
// MI455X (gfx1250) — compile-verified
//
#include <hip/hip_runtime.h>

typedef __attribute__((ext_vector_type(2))) float v2f;
typedef __attribute__((ext_vector_type(8))) float v8f;

#define D_FEAT 64
#define WT_STRIDE 68   // padded LDS row stride (floats): 8B aligned, bank-conflict-free

__global__ __launch_bounds__(256) void gcn_zero_f32(float* __restrict__ p, int n) {
  int i = blockIdx.x * blockDim.x + threadIdx.x;
  if (i < n) p[i] = 0.0f;
}

__global__ __launch_bounds__(256) void gcn_degrees(const int* __restrict__ src,
                                                   const int* __restrict__ dst,
                                                   float* __restrict__ outdeg,
                                                   float* __restrict__ indeg, int E) {
  int e = blockIdx.x * blockDim.x + threadIdx.x;
  if (e < E) {
    atomicAdd(&outdeg[src[e]], 1.0f);
    atomicAdd(&indeg[dst[e]], 1.0f);
  }
}

__global__ __launch_bounds__(256) void gcn_norms(float* __restrict__ outdeg,
                                                 float* __restrict__ indeg, int N) {
  int i = blockIdx.x * blockDim.x + threadIdx.x;
  if (i < N) {
    float od = outdeg[i];
    float id = indeg[i];
    outdeg[i] = (od > 0.0f) ? rsqrtf(od) : 0.0f;
    indeg[i]  = (id > 0.0f) ? rsqrtf(id) : 0.0f;
  }
}

// m = (h * norm_src)[src]; agg[dst] += m  -- 16 threads per edge, float4 each.
__global__ __launch_bounds__(256) void gcn_scatter(const float* __restrict__ h,
                                                   const float* __restrict__ norm_src,
                                                   const int* __restrict__ src,
                                                   const int* __restrict__ dst,
                                                   float* __restrict__ agg, int E) {
  int t = blockIdx.x * blockDim.x + threadIdx.x;
  int e = t >> 4;
  if (e >= E) return;
  int q = (t & 15) << 2;
  int s = src[e];
  int d = dst[e];
  float ns = norm_src[s];
  const float4 hv = *(const float4*)(h + (size_t)s * D_FEAT + q);
  float* base = agg + (size_t)d * D_FEAT + q;
  atomicAdd(base + 0, hv.x * ns);
  atomicAdd(base + 1, hv.y * ns);
  atomicAdd(base + 2, hv.z * ns);
  atomicAdd(base + 3, hv.w * ns);
}

// h_out = relu((agg * norm_dst) @ W + b) via V_WMMA_F32_16X16X4_F32.
// 8 waves/block, each wave owns a 16-row tile; whole W^T staged in LDS.
__global__ __launch_bounds__(256) void gcn_gemm_wmma(const float* __restrict__ agg,
                                                     const float* __restrict__ norm_dst,
                                                     const float* __restrict__ W,
                                                     const float* __restrict__ bias,
                                                     float* __restrict__ out, int n_rows) {
  __shared__ float lds[64 * WT_STRIDE + 8 * 16 * WT_STRIDE];
  float* Wt = lds;  // Wt[n][k] = W[k][n]
  const int tid = threadIdx.x;
  const int wave = tid >> 5;
  const int lane = tid & 31;

  // Stage W transposed (coalesced global read).
  for (int idx = tid; idx < 64 * 64; idx += 256) {
    int k = idx >> 6;
    int n = idx & 63;
    Wt[n * WT_STRIDE + k] = W[idx];
  }

  const int r0 = (blockIdx.x * 8 + wave) * 16;
  float* As = lds + 64 * WT_STRIDE + wave * 16 * WT_STRIDE;
  if (r0 < n_rows) {
    for (int idx = lane; idx < 16 * 64; idx += 32) {
      int row = idx >> 6;
      int col = idx & 63;
      int grow = r0 + row;
      float v = 0.0f;
      if (grow < n_rows) v = agg[(size_t)grow * D_FEAT + col] * norm_dst[grow];
      As[row * WT_STRIDE + col] = v;
    }
  }
  __syncthreads();
  if (r0 >= n_rows) return;

  const int half = lane >> 4;  // 0: K+{0,1}; 1: K+{2,3} per ISA f32 A/B fragment layout
  const int lr = lane & 15;
  const float* arow = As + lr * WT_STRIDE + 2 * half;
  const float* b0 = Wt + (0 + lr) * WT_STRIDE + 2 * half;
  const float* b1 = Wt + (16 + lr) * WT_STRIDE + 2 * half;
  const float* b2 = Wt + (32 + lr) * WT_STRIDE + 2 * half;
  const float* b3 = Wt + (48 + lr) * WT_STRIDE + 2 * half;

  v8f acc0 = {}, acc1 = {}, acc2 = {}, acc3 = {};
#pragma unroll
  for (int k0 = 0; k0 < 64; k0 += 4) {
    v2f a   = *(const v2f*)(arow + k0);
    v2f bb0 = *(const v2f*)(b0 + k0);
    v2f bb1 = *(const v2f*)(b1 + k0);
    v2f bb2 = *(const v2f*)(b2 + k0);
    v2f bb3 = *(const v2f*)(b3 + k0);
    acc0 = __builtin_amdgcn_wmma_f32_16x16x4_f32(false, a, false, bb0, (short)0, acc0, false, false);
    acc1 = __builtin_amdgcn_wmma_f32_16x16x4_f32(false, a, false, bb1, (short)0, acc1, false, false);
    acc2 = __builtin_amdgcn_wmma_f32_16x16x4_f32(false, a, false, bb2, (short)0, acc2, false, false);
    acc3 = __builtin_amdgcn_wmma_f32_16x16x4_f32(false, a, false, bb3, (short)0, acc3, false, false);
  }

  float bv0 = bias[0 + lr];
  float bv1 = bias[16 + lr];
  float bv2 = bias[32 + lr];
  float bv3 = bias[48 + lr];

#pragma unroll
  for (int j = 0; j < 8; ++j) {
    int row = r0 + j + 8 * half;  // C/D layout: VGPR j -> M=j (lanes 0-15) / M=j+8 (16-31)
    if (row < n_rows) {
      float* o = out + (size_t)row * D_FEAT;
      o[0 + lr]  = fmaxf(acc0[j] + bv0, 0.0f);
      o[16 + lr] = fmaxf(acc1[j] + bv1, 0.0f);
      o[32 + lr] = fmaxf(acc2[j] + bv2, 0.0f);
      o[48 + lr] = fmaxf(acc3[j] + bv3, 0.0f);
    }
  }
}

extern "C" void kernel_launch(void* const* d_in, const int* in_sizes, int n_in,
                              void* d_out, int out_size, void* d_ws, size_t ws_size,
                              hipStream_t stream) {
  const float* x  = (const float*)d_in[0];
  const int*   ei = (const int*)d_in[1];
  const float* Ws = (const float*)d_in[2];
  const float* bs = (const float*)d_in[3];
  const int N = in_sizes[0] / D_FEAT;
  const int E = in_sizes[1] / 2;
  const int* src = ei;       // edge_index[0]
  const int* dst = ei + E;   // edge_index[1]

  float* out = (float*)d_out;
  float* ws = (float*)d_ws;
  float* normsrc = ws;                             // N  (out_deg -> norm_src)
  float* normdst = ws + (size_t)N;                 // N  (in_deg  -> norm_dst)
  float* agg     = ws + 2 * (size_t)N;             // N*64
  float* hmid    = agg + (size_t)N * D_FEAT;       // N*64

  const int B = 256;
  // Degrees + norms
  gcn_zero_f32<<<(2 * N + B - 1) / B, B, 0, stream>>>(normsrc, 2 * N);
  gcn_degrees<<<(E + B - 1) / B, B, 0, stream>>>(src, dst, normsrc, normdst, E);
  gcn_norms<<<(N + B - 1) / B, B, 0, stream>>>(normsrc, normdst, N);

  const float* hin = x;
  for (int l = 0; l < 3; ++l) {
    float* hout = (l == 1) ? hmid : out;  // L0->out, L1->ws, L2->out
    gcn_zero_f32<<<((N * D_FEAT) + B - 1) / B, B, 0, stream>>>(agg, N * D_FEAT);
    gcn_scatter<<<(int)(((size_t)E * 16 + B - 1) / B), B, 0, stream>>>(hin, normsrc, src, dst, agg, E);
    gcn_gemm_wmma<<<(N + 127) / 128, B, 0, stream>>>(agg, normdst,
                                                     Ws + (size_t)l * D_FEAT * D_FEAT,
                                                     bs + (size_t)l * D_FEAT, hout, N);
    hin = hout;
  }
}